// dilated_involution3_82274393522458
// MI455X (gfx1250) — compile-verified
//
#include <hip/hip_runtime.h>
#include <hip/hip_bf16.h>

// CDNA5 (gfx1250) fused dilated involution.
// WMMA f32 16x16x4 layout conventions (ISA 7.12.2, wave32):
//   A (16x4, MxK): lane L holds M=L%16; VGPR v holds K = 2*(L/16) + v
//   B (4x16, KxN): lane L holds N=L%16; VGPR v holds K = 2*(L/16) + v
//   C/D (16x16) : lane L holds N=L%16; VGPR v holds M = v + 8*(L/16)

typedef __attribute__((ext_vector_type(2))) float v2f;
typedef __attribute__((ext_vector_type(8))) float v8f;

#define HW    16384   // 128*128
#define CRED  16
#define EPS   1e-5f

// ---------------- kernel 0: zero the stats slab (harness poisons ws) --------
__global__ void k0_zero(float* __restrict__ stats) {
    stats[threadIdx.x] = 0.0f;   // 256 floats: sum[128], sumsq[128]
}

// ---------------- kernel 1: t_raw = w1 @ x  (+ per-(b,o) sum / sumsq) -------
// grid = B*H blocks, 256 threads (8 waves); wave w handles pixels [16w,16w+16)
__global__ __launch_bounds__(256) void k1_conv1_stats(
    const float* __restrict__ x,      // (8,128,128,128)
    const float* __restrict__ w1,     // (16,128) row-major
    float* __restrict__ t_raw,        // (8,16,128,128)
    float* __restrict__ stats)        // sum[128], sumsq[128]
{
    __shared__ float s_sum[CRED];
    __shared__ float s_sq[CRED];
    const int tid = threadIdx.x;
    if (tid < CRED) { s_sum[tid] = 0.0f; s_sq[tid] = 0.0f; }
    __syncthreads();

    const int b    = blockIdx.x >> 7;
    const int y    = blockIdx.x & 127;
    const int lane = tid & 31;
    const int n    = lane & 15;      // N index (pixel) / M index (A row)
    const int hi   = lane >> 4;      // lane-half
    const int x0   = (tid >> 5) * 16;

    const float* xb = x + (size_t)b * 128 * HW + y * 128 + x0 + n;

    v8f acc = {};
    #pragma unroll 4
    for (int s = 0; s < 32; ++s) {          // K = 128 in steps of 4
        const int k0 = 4 * s + 2 * hi;
        v2f a, bm;
        a[0]  = w1[n * 128 + k0];
        a[1]  = w1[n * 128 + k0 + 1];
        bm[0] = xb[(size_t)k0 * HW];
        bm[1] = xb[(size_t)(k0 + 1) * HW];
        acc = __builtin_amdgcn_wmma_f32_16x16x4_f32(
                  false, a, false, bm, (short)0, acc, false, false);
    }

    // write t_raw + reduce stats (16-lane halves via xor shuffles)
    float* trow = t_raw + (size_t)b * CRED * HW + y * 128 + x0 + n;
    #pragma unroll
    for (int v = 0; v < 8; ++v) {
        const int ch = v + 8 * hi;          // channel this D row holds
        const float d = acc[v];
        trow[(size_t)ch * HW] = d;
        float sm = d, sq = d * d;
        #pragma unroll
        for (int off = 8; off >= 1; off >>= 1) {  // xor<16: stays in half
            sm += __shfl_xor(sm, off, 32);
            sq += __shfl_xor(sq, off, 32);
        }
        if (n == 0) {
            atomicAdd(&s_sum[ch], sm);
            atomicAdd(&s_sq[ch], sq);
        }
    }
    __syncthreads();
    if (tid < CRED) {
        atomicAdd(&stats[b * CRED + tid], s_sum[tid]);
        atomicAdd(&stats[128 + b * CRED + tid], s_sq[tid]);
    }
}

// ---------------- kernel 2: fused IN/ReLU + conv2(WMMA) + unfold + reduce ---
// grid = B*H blocks, 256 threads (8 waves); block owns one (b, y) output row.
// Dynamic LDS: accum[16384] | w2A[18432] | scale[16] | shift[16]
__global__ __launch_bounds__(256) void k2_involution(
    const float* __restrict__ x,
    const float* __restrict__ t_raw,
    const float* __restrict__ stats,
    const float* __restrict__ gamma,
    const float* __restrict__ beta,
    const float* __restrict__ w2,     // (1152,16) row-major
    const float* __restrict__ b2,     // (1152)
    float* __restrict__ out)
{
    extern __shared__ float lds[];
    float* accum   = lds;                   // 16384 f (64 KB) out-row accum
    float* w2A     = lds + HW;              // 18432 f, WMMA-A layout
    float* s_scale = lds + HW + 18432;      // 16 f
    float* s_shift = s_scale + CRED;        // 16 f

    const int tid = threadIdx.x;
    const int b   = blockIdx.x >> 7;
    const int y   = blockIdx.x & 127;

    // fold InstanceNorm affine into per-channel scale/shift
    if (tid < CRED) {
        const float inv  = 1.0f / (float)HW;
        const float sm   = stats[b * CRED + tid];
        const float sq   = stats[128 + b * CRED + tid];
        const float mean = sm * inv;
        const float var  = sq * inv - mean * mean;
        const float rstd = rsqrtf(var + EPS);
        const float sc   = gamma[tid] * rstd;
        s_scale[tid] = sc;
        s_shift[tid] = beta[tid] - mean * sc;
    }
    for (int idx = tid; idx < HW; idx += 256) accum[idx] = 0.0f;
    // stage w2 into LDS pre-swizzled: chunk (m,s) -> 64 floats, lane L owns 2
    for (int idx = tid; idx < 18432; idx += 256) {
        const int chunk = idx >> 6;          // m*4 + s
        const int rem   = idx & 63;
        const int L     = rem >> 1;
        const int v     = rem & 1;
        const int m     = chunk >> 2;
        const int s     = chunk & 3;
        const int row   = m * 16 + (L & 15);
        const int k     = 4 * s + 2 * (L >> 4) + v;
        w2A[idx] = w2[row * 16 + k];
    }
    __syncthreads();

    const int lane = tid & 31;
    const int n    = lane & 15;
    const int hi   = lane >> 4;
    const int x0   = (tid >> 5) * 16;

    // B operand: normalized + ReLU'd t for this wave's 16 pixels (K=16)
    v2f Bc[4];
    const float* tb = t_raw + (size_t)b * CRED * HW + y * 128 + x0 + n;
    #pragma unroll
    for (int s = 0; s < 4; ++s) {
        #pragma unroll
        for (int v = 0; v < 2; ++v) {
            const int k  = 4 * s + 2 * hi + v;
            const float t = tb[(size_t)k * HW] * s_scale[k] + s_shift[k];
            Bc[s][v] = fmaxf(t, 0.0f);
        }
    }

    const float* xb = x + (size_t)b * 128 * HW;

    for (int m = 0; m < 72; ++m) {           // 72 M-tiles cover 1152 rows
        v8f acc = {};
        #pragma unroll
        for (int s = 0; s < 4; ++s) {        // ds_load_b64, conflict-free
            const v2f a = *(const v2f*)(w2A + (size_t)(m * 4 + s) * 64 + lane * 2);
            acc = __builtin_amdgcn_wmma_f32_16x16x4_f32(
                      false, a, false, Bc[s], (short)0, acc, false, false);
        }
        #pragma unroll
        for (int v = 0; v < 8; ++v) {
            const unsigned r   = (unsigned)(m * 16 + v + 8 * hi); // w2 row
            const float    wv  = acc[v] + b2[r];
            const unsigned c   = r / 9u;       // channel
            const unsigned tap = r - c * 9u;   // kernel tap
            const int ti = (int)(tap / 3u);
            const int tj = (int)(tap - 3u * (tap / 3u));
            const int py = y - 2 + 2 * ti;               // uniform per half
            const int px = x0 + n - 2 + 2 * tj;          // contiguous in lane
            float xv = 0.0f;
            if ((unsigned)py < 128u && (unsigned)px < 128u)
                xv = xb[(size_t)c * HW + py * 128 + px];
            atomicAdd(&accum[c * 128 + x0 + n], wv * xv); // ds_add_f32
        }
    }
    __syncthreads();

    float* ob = out + (size_t)b * 128 * HW + y * 128;
    for (int idx = tid; idx < HW; idx += 256) {
        const int c  = idx >> 7;
        const int xx = idx & 127;
        ob[(size_t)c * HW + xx] = accum[idx];
    }
}

extern "C" void kernel_launch(void* const* d_in, const int* in_sizes, int n_in,
                              void* d_out, int out_size, void* d_ws, size_t ws_size,
                              hipStream_t stream) {
    const float* x     = (const float*)d_in[0];
    const float* w1    = (const float*)d_in[1];
    const float* gamma = (const float*)d_in[2];
    const float* beta  = (const float*)d_in[3];
    const float* w2    = (const float*)d_in[4];
    const float* b2    = (const float*)d_in[5];
    float* out = (float*)d_out;

    float* stats = (float*)d_ws;                       // 256 floats
    float* t_raw = (float*)((char*)d_ws + 4096);       // 8*16*128*128 floats (8 MB)

    k0_zero<<<1, 256, 0, stream>>>(stats);
    k1_conv1_stats<<<1024, 256, 0, stream>>>(x, w1, t_raw, stats);

    const size_t ldsBytes = (size_t)(HW + 18432 + 2 * CRED) * sizeof(float); // ~139 KB
    k2_involution<<<1024, 256, ldsBytes, stream>>>(x, t_raw, stats, gamma, beta,
                                                   w2, b2, out);
}